// DecoderRNN_17394617548943
// MI455X (gfx1250) — compile-verified
//
#include <hip/hip_runtime.h>

typedef __attribute__((ext_vector_type(2))) float v2f;
typedef __attribute__((ext_vector_type(8))) float v8f;

#define BB     16
#define TT     600
#define EE     512
#define HH     512
#define AD     512
#define VV     8000
#define GATES  1536
#define XDIM   1024
#define STEPS  64

// ---- WMMA helpers (f32 16x16x4, one wave computes one 16x16 tile) ----------
// A: 16x4 f32  -> 2 VGPR/lane: lane<16 holds {A[r][k],A[r][k+1]}, lane>=16 {A[r][k+2],A[r][k+3]}
// B: 4x16 f32  -> symmetric striping, col = lane&15
// C/D: 16x16   -> VGPR i: lanes 0-15 M=i, lanes 16-31 M=i+8

__device__ __forceinline__ v8f wmma_k(const float* __restrict__ Arow,
                                      const float* __restrict__ Brow,
                                      int K, v8f acc, int koff) {
  for (int k = 0; k < K; k += 4) {
    v2f a, b;
    a.x = Arow[k + koff];
    a.y = Arow[k + koff + 1];
    b.x = Brow[k + koff];
    b.y = Brow[k + koff + 1];
    acc = __builtin_amdgcn_wmma_f32_16x16x4_f32(false, a, false, b,
                                                (short)0, acc, false, false);
  }
  return acc;
}

__device__ __forceinline__ void store_tile(float* __restrict__ out, int ldo,
                                           int m0, int n0, v8f acc,
                                           const float* __restrict__ bias) {
  int lane = threadIdx.x & 31;
  int col  = lane & 15;
  int mhi  = (lane >> 4) * 8;
  float bv = bias ? bias[n0 + col] : 0.0f;
#pragma unroll
  for (int i = 0; i < 8; ++i)
    out[(size_t)(m0 + mhi + i) * ldo + n0 + col] = acc[i] + bv;
}

// ---- init recurrent state ---------------------------------------------------
__global__ void k_init(float* h, float* ctx, float* aw, int* tok) {
  int i = blockIdx.x * blockDim.x + threadIdx.x;
  if (i < BB * HH) h[i] = 0.f;
  if (i < BB * EE) ctx[i] = 0.f;
  if (i < BB * TT) aw[i] = 0.f;
  if (i < BB)      tok[i] = 1;  // SOS
}

// ---- Vv = enc @ attn_V^T  [9600 x 512] -------------------------------------
__global__ void k_vv(const float* __restrict__ enc,
                     const float* __restrict__ attn_V,
                     float* __restrict__ Vv) {
  int wave = (blockIdx.x * blockDim.x + threadIdx.x) >> 5;
  const int NT = AD / 16;                 // 32
  int mt = wave / NT, nt = wave % NT;
  if (mt >= (BB * TT) / 16) return;       // 600 m-tiles (uniform per wave)
  int lane = threadIdx.x & 31;
  int r = lane & 15, koff = (lane >> 4) * 2;
  v8f acc = {};
  acc = wmma_k(enc + (size_t)(mt * 16 + r) * EE,
               attn_V + (size_t)(nt * 16 + r) * EE, EE, acc, koff);
  store_tile(Vv, AD, mt * 16, nt * 16, acc, nullptr);
}

// ---- x = [emb[tok], ctx] ----------------------------------------------------
__global__ void k_embed(const float* __restrict__ emb, const int* __restrict__ tok,
                        const float* __restrict__ ctx, float* __restrict__ x) {
  int i = blockIdx.x * blockDim.x + threadIdx.x;
  if (i >= BB * XDIM) return;
  int m = i / XDIM, k = i % XDIM;
  x[i] = (k < HH) ? emb[(size_t)tok[m] * HH + k] : ctx[m * EE + (k - HH)];
}

// ---- GRU gate GEMMs: G1 = x@w_ih^T + b_ih, G2 = h@w_hh^T + b_hh ------------
__global__ void k_gates(const float* __restrict__ x, const float* __restrict__ h,
                        const float* __restrict__ w_ih, const float* __restrict__ w_hh,
                        const float* __restrict__ b_ih, const float* __restrict__ b_hh,
                        float* __restrict__ G1, float* __restrict__ G2) {
  int wave = (blockIdx.x * blockDim.x + threadIdx.x) >> 5;
  int lane = threadIdx.x & 31;
  int r = lane & 15, koff = (lane >> 4) * 2;
  v8f acc = {};
  const int NT = GATES / 16;              // 96
  if (wave < NT) {
    int n0 = wave * 16;
    acc = wmma_k(x + (size_t)r * XDIM, w_ih + (size_t)(n0 + r) * XDIM, XDIM, acc, koff);
    store_tile(G1, GATES, 0, n0, acc, b_ih);
  } else if (wave < 2 * NT) {
    int n0 = (wave - NT) * 16;
    acc = wmma_k(h + (size_t)r * HH, w_hh + (size_t)(n0 + r) * HH, HH, acc, koff);
    store_tile(G2, GATES, 0, n0, acc, b_hh);
  }
}

// ---- GRU nonlinearity (PyTorch gate order r,z,n) ---------------------------
__global__ void k_hnew(const float* __restrict__ G1, const float* __restrict__ G2,
                       float* __restrict__ h, float* __restrict__ xo) {
  int i = blockIdx.x * blockDim.x + threadIdx.x;
  if (i >= BB * HH) return;
  int m = i / HH, j = i % HH;
  float xr = G1[m * GATES + j],          hr = G2[m * GATES + j];
  float xz = G1[m * GATES + HH + j],     hz = G2[m * GATES + HH + j];
  float xn = G1[m * GATES + 2 * HH + j], hn = G2[m * GATES + 2 * HH + j];
  float rg = 1.f / (1.f + __expf(-(xr + hr)));
  float zg = 1.f / (1.f + __expf(-(xz + hz)));
  float ng = tanhf(xn + rg * hn);
  float hv = (1.f - zg) * ng + zg * h[i];
  h[i] = hv;
  xo[m * XDIM + j] = hv;
}

// ---- wq = h_new @ attn_W^T --------------------------------------------------
__global__ void k_wq(const float* __restrict__ h, const float* __restrict__ attn_W,
                     float* __restrict__ wq) {
  int wave = (blockIdx.x * blockDim.x + threadIdx.x) >> 5;
  if (wave >= AD / 16) return;
  int lane = threadIdx.x & 31;
  int r = lane & 15, koff = (lane >> 4) * 2;
  v8f acc = {};
  int n0 = wave * 16;
  acc = wmma_k(h + (size_t)r * HH, attn_W + (size_t)(n0 + r) * HH, HH, acc, koff);
  store_tile(wq, AD, 0, n0, acc, nullptr);
}

// ---- score[b,t] = fc( tanh(wq + Vv + conv(aw) + attn_b) ) ------------------
__global__ void k_score(const float* __restrict__ wq, const float* __restrict__ Vv,
                        const float* __restrict__ aw,
                        const float* __restrict__ conv_w, const float* __restrict__ conv_b,
                        const float* __restrict__ attn_b,
                        const float* __restrict__ fcw, const float* __restrict__ fcb,
                        float* __restrict__ score) {
  int wave = (blockIdx.x * blockDim.x + threadIdx.x) >> 5;
  if (wave >= BB * TT) return;
  int lane = threadIdx.x & 31;
  int b = wave / TT, t = wave % TT;
  float am1 = (t > 0)      ? aw[b * TT + t - 1] : 0.f;
  float a0  = aw[b * TT + t];
  float ap1 = (t < TT - 1) ? aw[b * TT + t + 1] : 0.f;
  float acc = 0.f;
  for (int a = lane; a < AD; a += 32) {
    float cv = conv_b[a] + conv_w[a * 3] * am1 + conv_w[a * 3 + 1] * a0 +
               conv_w[a * 3 + 2] * ap1;
    float e = tanhf(wq[b * AD + a] + Vv[((size_t)b * TT + t) * AD + a] + cv + attn_b[a]);
    acc += e * fcw[a];
  }
  for (int off = 16; off > 0; off >>= 1) acc += __shfl_down(acc, off, 32);
  if (lane == 0) score[b * TT + t] = acc + fcb[0];
}

// ---- softmax over T, context = aw @ enc ------------------------------------
__global__ void k_softmax_ctx(const float* __restrict__ score,
                              const float* __restrict__ enc,
                              float* __restrict__ aw, float* __restrict__ ctx,
                              float* __restrict__ xo) {
  __shared__ float red[256];
  __shared__ float s_aw[TT];
  int b = blockIdx.x, tid = threadIdx.x;
  float m = -3.4e38f;
  for (int t = tid; t < TT; t += 256) m = fmaxf(m, score[b * TT + t]);
  red[tid] = m; __syncthreads();
  for (int s = 128; s > 0; s >>= 1) {
    if (tid < s) red[tid] = fmaxf(red[tid], red[tid + s]);
    __syncthreads();
  }
  float mx = red[0]; __syncthreads();
  float sum = 0.f;
  for (int t = tid; t < TT; t += 256) {
    float e = __expf(score[b * TT + t] - mx);
    s_aw[t] = e; sum += e;
  }
  red[tid] = sum; __syncthreads();
  for (int s = 128; s > 0; s >>= 1) {
    if (tid < s) red[tid] += red[tid + s];
    __syncthreads();
  }
  float inv = 1.f / red[0]; __syncthreads();
  for (int t = tid; t < TT; t += 256) { s_aw[t] *= inv; aw[b * TT + t] = s_aw[t]; }
  __syncthreads();
  for (int d = tid; d < EE; d += 256) {
    float c = 0.f;
    for (int t = 0; t < TT; ++t) c += s_aw[t] * enc[((size_t)b * TT + t) * EE + d];
    ctx[b * EE + d] = c;
    xo[b * XDIM + HH + d] = c;
  }
}

// ---- logits = [h,ctx] @ fc_w^T + fc_b --------------------------------------
__global__ void k_logits(const float* __restrict__ xo, const float* __restrict__ fc_w,
                         const float* __restrict__ fc_b, float* __restrict__ logits) {
  int wave = (blockIdx.x * blockDim.x + threadIdx.x) >> 5;
  if (wave >= VV / 16) return;
  int lane = threadIdx.x & 31;
  int r = lane & 15, koff = (lane >> 4) * 2;
  v8f acc = {};
  int n0 = wave * 16;
  acc = wmma_k(xo + (size_t)r * XDIM, fc_w + (size_t)(n0 + r) * XDIM, XDIM, acc, koff);
  store_tile(logits, VV, 0, n0, acc, fc_b);
}

// ---- log_softmax over vocab, greedy argmax ---------------------------------
__global__ void k_logsoftmax(const float* __restrict__ logits, float* __restrict__ out,
                             int* __restrict__ tok, int step) {
  __shared__ float red[256];
  __shared__ int   redi[256];
  int b = blockIdx.x, tid = threadIdx.x;
  float m = -3.4e38f; int mi = 0;
  for (int v = tid; v < VV; v += 256) {
    float l = logits[(size_t)b * VV + v];
    if (l > m) { m = l; mi = v; }
  }
  red[tid] = m; redi[tid] = mi; __syncthreads();
  for (int s = 128; s > 0; s >>= 1) {
    if (tid < s) {
      if (red[tid + s] > red[tid] ||
          (red[tid + s] == red[tid] && redi[tid + s] < redi[tid])) {
        red[tid] = red[tid + s]; redi[tid] = redi[tid + s];
      }
    }
    __syncthreads();
  }
  float mx = red[0]; int amax = redi[0]; __syncthreads();
  float sum = 0.f;
  for (int v = tid; v < VV; v += 256) sum += __expf(logits[(size_t)b * VV + v] - mx);
  red[tid] = sum; __syncthreads();
  for (int s = 128; s > 0; s >>= 1) {
    if (tid < s) red[tid] += red[tid + s];
    __syncthreads();
  }
  float lse = mx + __logf(red[0]);
  for (int v = tid; v < VV; v += 256)
    out[((size_t)b * STEPS + step) * VV + v] = logits[(size_t)b * VV + v] - lse;
  if (tid == 0) tok[b] = amax;
}

// ---- host -------------------------------------------------------------------
extern "C" void kernel_launch(void* const* d_in, const int* in_sizes, int n_in,
                              void* d_out, int out_size, void* d_ws, size_t ws_size,
                              hipStream_t stream) {
  const float* enc      = (const float*)d_in[0];
  const float* emb      = (const float*)d_in[1];
  const float* w_ih     = (const float*)d_in[2];
  const float* w_hh     = (const float*)d_in[3];
  const float* b_ih     = (const float*)d_in[4];
  const float* b_hh     = (const float*)d_in[5];
  const float* conv_w   = (const float*)d_in[6];
  const float* conv_b   = (const float*)d_in[7];
  const float* attn_W   = (const float*)d_in[8];
  const float* attn_V   = (const float*)d_in[9];
  const float* attn_fcw = (const float*)d_in[10];
  const float* attn_fcb = (const float*)d_in[11];
  const float* attn_b   = (const float*)d_in[12];
  const float* fc_w     = (const float*)d_in[13];
  const float* fc_b     = (const float*)d_in[14];
  float* out = (float*)d_out;

  float* ws = (float*)d_ws;
  size_t off = 0;
  auto alloc = [&](size_t n) { float* p = ws + off; off += n; return p; };
  float* Vv    = alloc((size_t)BB * TT * AD);
  float* x     = alloc((size_t)BB * XDIM);
  float* G1    = alloc((size_t)BB * GATES);
  float* G2    = alloc((size_t)BB * GATES);
  float* wq    = alloc((size_t)BB * AD);
  float* score = alloc((size_t)BB * TT);
  float* h     = alloc((size_t)BB * HH);
  float* ctx   = alloc((size_t)BB * EE);
  float* aw    = alloc((size_t)BB * TT);
  float* xo    = alloc((size_t)BB * XDIM);
  float* logit = alloc((size_t)BB * VV);
  int*   tok   = (int*)alloc(BB);

  k_init<<<(BB * TT + 255) / 256, 256, 0, stream>>>(h, ctx, aw, tok);
  {
    int waves = (BB * TT / 16) * (AD / 16);     // 19200 tiles
    k_vv<<<(waves + 7) / 8, 256, 0, stream>>>(enc, attn_V, Vv);
  }
  for (int step = 0; step < STEPS; ++step) {
    k_embed<<<(BB * XDIM + 255) / 256, 256, 0, stream>>>(emb, tok, ctx, x);
    k_gates<<<(2 * (GATES / 16) + 7) / 8, 256, 0, stream>>>(x, h, w_ih, w_hh,
                                                            b_ih, b_hh, G1, G2);
    k_hnew<<<(BB * HH + 255) / 256, 256, 0, stream>>>(G1, G2, h, xo);
    k_wq<<<((AD / 16) + 7) / 8, 256, 0, stream>>>(h, attn_W, wq);
    k_score<<<(BB * TT + 7) / 8, 256, 0, stream>>>(wq, Vv, aw, conv_w, conv_b,
                                                   attn_b, attn_fcw, attn_fcb, score);
    k_softmax_ctx<<<BB, 256, 0, stream>>>(score, enc, aw, ctx, xo);
    k_logits<<<((VV / 16) + 7) / 8, 256, 0, stream>>>(xo, fc_w, fc_b, logit);
    k_logsoftmax<<<BB, 256, 0, stream>>>(logit, out, tok, step);
  }
}